// TripletLoss_33758442946983
// MI455X (gfx1250) — compile-verified
//
#include <hip/hip_runtime.h>

#define NPTS 8192
#define DDIM 512
#define TMARGIN 0.2f

// K-chunk staged per TDM transfer, LDS row stride padded 32 -> 34 dwords
// (TDM pad: pad_interval=4 (32 dwords), pad_amount=1 (2 dwords)).
#define KC 32
#define LDS_STRIDE 34
#define SZA (128 * LDS_STRIDE)          /* dwords: A tile 128 rows   */
#define SZB (256 * LDS_STRIDE)          /* dwords: B tile 256 rows   */
#define SZBUF (SZA + SZB)               /* dwords per buffer = 13056 */
#define LDS_BYTES (SZBUF * 2 * 4)       /* 104448 B double-buffered  */

typedef __attribute__((ext_vector_type(2))) float v2f;
typedef __attribute__((ext_vector_type(8))) float v8f;
typedef __attribute__((ext_vector_type(4))) unsigned int v4u;
typedef __attribute__((ext_vector_type(8))) int v8i;
typedef __attribute__((ext_vector_type(4))) int v4i;

// Monotonic float <-> uint mapping so atomicMin(u32) == float min (handles negatives).
__device__ __forceinline__ unsigned f2ord(float f) {
    unsigned u = __float_as_uint(f);
    return (u & 0x80000000u) ? ~u : (u | 0x80000000u);
}
__device__ __forceinline__ float ord2f(unsigned o) {
    unsigned u = (o & 0x80000000u) ? (o & 0x7FFFFFFFu) : ~o;
    return __uint_as_float(u);
}

// Issue one TDM 2-D tile load: tileRows x KC fp32 elements, row stride DDIM,
// into LDS at ldsOff with 2-dword padding after every 32-dword row (stride 34).
// D# packing per CDNA5 ISA 8.3/8.4 (count=1, type=2, data_size=4B, pad on).
__device__ __forceinline__ void tdm_load_tile(unsigned ldsOff, const float* gptr,
                                              int tileRows) {
    unsigned long long ga = (unsigned long long)(size_t)gptr;
    v4u g0;
    g0[0] = 1u;                                            // count=1, user desc
    g0[1] = ldsOff;                                        // lds_addr [63:32]
    g0[2] = (unsigned)(ga & 0xFFFFFFFFull);                // global_addr[31:0]
    g0[3] = (unsigned)((ga >> 32) & 0x1FFFFFFull)          // global_addr[56:32]
            | (2u << 30);                                  // type=2 (image)
    v8i g1;
    g1[0] = (int)((2u << 16)      /* data_size=4B      */
                | (1u << 20)      /* pad_enable        */
                | (4u << 22)      /* pad_interval: 32dw*/
                | (1u << 25));    /* pad_amount: 2dw   */
    g1[1] = (int)(((unsigned)DDIM & 0xFFFFu) << 16);       // tensor_dim0 lo16
    g1[2] = (int)((((unsigned)DDIM >> 16) & 0xFFFFu)
                | (((unsigned)NPTS & 0xFFFFu) << 16));     // {t_dim1 lo16, t_dim0 hi16}
    g1[3] = (int)((((unsigned)NPTS >> 16) & 0xFFFFu)
                | ((unsigned)KC << 16));                   // {tile_dim0, t_dim1 hi16}
    g1[4] = (int)((unsigned)tileRows & 0xFFFFu);           // {tile_dim2=0, tile_dim1}
    g1[5] = (int)DDIM;                                     // tensor_dim0_stride lo32
    g1[6] = 0;                                             // strides hi bits
    g1[7] = 0;
    v4i z4 = {0, 0, 0, 0};                                 // 2-D tile: groups 2/3 unused
    v8i z8 = {0, 0, 0, 0, 0, 0, 0, 0};                     // unused extended group
    __builtin_amdgcn_tensor_load_to_lds(g0, g1, z4, z4, z8, 0);
}

// One wave per row: a2 = ||anchor||^2, b2 = ||negative||^2, dap = ||anchor-positive||^2,
// and initialize the per-row min accumulators (workspace is poisoned by the harness).
__global__ __launch_bounds__(256) void tl_prep(
    const float* __restrict__ anchor, const float* __restrict__ positive,
    const float* __restrict__ negative,
    float* __restrict__ a2, float* __restrict__ b2, float* __restrict__ dap,
    unsigned* __restrict__ minSemi, unsigned* __restrict__ minValid)
{
    const int wave = threadIdx.x >> 5;
    const int lane = threadIdx.x & 31;
    const int row  = blockIdx.x * 8 + wave;
    const float4* ar = (const float4*)(anchor   + (size_t)row * DDIM);
    const float4* pr = (const float4*)(positive + (size_t)row * DDIM);
    const float4* nr = (const float4*)(negative + (size_t)row * DDIM);
    float sa = 0.f, sb = 0.f, sd = 0.f;
    #pragma unroll
    for (int t = lane; t < DDIM / 4; t += 32) {
        float4 a = ar[t], p = pr[t], n = nr[t];
        sa += a.x*a.x + a.y*a.y + a.z*a.z + a.w*a.w;
        sb += n.x*n.x + n.y*n.y + n.z*n.z + n.w*n.w;
        float dx = a.x - p.x, dy = a.y - p.y, dz = a.z - p.z, dw = a.w - p.w;
        sd += dx*dx + dy*dy + dz*dz + dw*dw;
    }
    #pragma unroll
    for (int off = 16; off; off >>= 1) {
        sa += __shfl_xor(sa, off, 32);
        sb += __shfl_xor(sb, off, 32);
        sd += __shfl_xor(sd, off, 32);
    }
    if (lane == 0) {
        a2[row] = sa; b2[row] = sb; dap[row] = sd;
        minSemi[row]  = 0xFFFFFFFFu;
        minValid[row] = 0xFFFFFFFFu;
    }
}

// Fused GEMM (anchor @ negative^T via v_wmma_f32_16x16x4_f32) + mask + per-row mins.
// TDM double-buffers 128x32 (A) and 256x32 (B) fp32 tiles into padded LDS; each of
// the 8 waves computes a 64x64 tile of the 128x256 block tile from bank-conflict-free
// ds_load_b64 fragment reads.
__global__ __launch_bounds__(256) void tl_pairmin(
    const float* __restrict__ anchor, const float* __restrict__ negative,
    const float* __restrict__ a2, const float* __restrict__ b2,
    const float* __restrict__ dap,
    const int* __restrict__ tA, const int* __restrict__ tN,
    unsigned* __restrict__ minSemi, unsigned* __restrict__ minValid)
{
    extern __shared__ float smem[];
    const int waveId = threadIdx.x >> 5;
    const int lane   = threadIdx.x & 31;
    const int hf     = lane >> 4;       // half-wave selects K+2 / M+8 per ISA layouts
    const int l16    = lane & 15;
    const int rowBase = blockIdx.x * 128 + (waveId >> 2) * 64;
    const int colBase = blockIdx.y * 256 + (waveId & 3) * 64;

    const unsigned ldsOff = (unsigned)(size_t)smem;        // LDS byte offset
    const float* gA = anchor   + (size_t)blockIdx.x * 128 * DDIM;
    const float* gB = negative + (size_t)blockIdx.y * 256 * DDIM;

    v8f acc[4][4];
    #pragma unroll
    for (int i = 0; i < 4; ++i)
        #pragma unroll
        for (int j = 0; j < 4; ++j)
            #pragma unroll
            for (int e = 0; e < 8; ++e) acc[i][j][e] = 0.0f;

    // Per-lane fragment dword indices into a buffer (A frag: lane=M, pair {K,K+1},
    // half-wave adds K+2; B frag mirrors with lane=N).
    int aIdx[4], bIdx[4];
    #pragma unroll
    for (int t = 0; t < 4; ++t) {
        aIdx[t] = ((waveId >> 2) * 64 + t * 16 + l16) * LDS_STRIDE + 2 * hf;
        bIdx[t] = SZA + ((waveId & 3) * 64 + t * 16 + l16) * LDS_STRIDE + 2 * hf;
    }

    const int NC = DDIM / KC;   // 16 K-chunks
    if (waveId == 0) {
        tdm_load_tile(ldsOff,           gA, 128);
        tdm_load_tile(ldsOff + SZA * 4, gB, 256);
    }
    for (int c = 0; c < NC; ++c) {
        const int cur = c & 1;
        if (waveId == 0) {
            if (c + 1 < NC) {
                const unsigned nb = ldsOff + (unsigned)((c + 1) & 1) * SZBUF * 4;
                tdm_load_tile(nb,           gA + (c + 1) * KC, 128);
                tdm_load_tile(nb + SZA * 4, gB + (c + 1) * KC, 256);
                __builtin_amdgcn_s_wait_tensorcnt(2);  // chunk c's two DMAs done
            } else {
                __builtin_amdgcn_s_wait_tensorcnt(0);
            }
        }
        __syncthreads();                                   // publish chunk c
        const float* buf = smem + cur * SZBUF;
        #pragma unroll
        for (int kk = 0; kk < KC; kk += 4) {
            v2f af[4], bf[4];
            #pragma unroll
            for (int t = 0; t < 4; ++t) {
                af[t] = *(const v2f*)(buf + aIdx[t] + kk);
                bf[t] = *(const v2f*)(buf + bIdx[t] + kk);
            }
            #pragma unroll
            for (int i = 0; i < 4; ++i)
                #pragma unroll
                for (int j = 0; j < 4; ++j)
                    acc[i][j] = __builtin_amdgcn_wmma_f32_16x16x4_f32(
                        false, af[i], false, bf[j], (short)0, acc[i][j], false, false);
        }
        __syncthreads();                                   // all done reading buf[cur]
    }

    // Epilogue: dist = a2[row] + b2[col] - 2*dot, masks per reference, per-row mins.
    float b2c[4], dapc[4];
    int   tnc[4];
    #pragma unroll
    for (int j = 0; j < 4; ++j) {
        int col = colBase + j * 16 + l16;
        b2c[j]  = b2[col];
        dapc[j] = dap[col];   // NOTE: reference compares against column-indexed d_ap[j]
        tnc[j]  = tN[col];
    }
    const float INF = __builtin_inff();
    #pragma unroll
    for (int i = 0; i < 4; ++i) {
        #pragma unroll
        for (int r = 0; r < 8; ++r) {
            // C/D layout: VGPR r, lanes 0-15 -> M=r, lanes 16-31 -> M=r+8
            int row = rowBase + i * 16 + r + 8 * hf;
            float a2r = a2[row];
            int   tar = tA[row];
            float mv = INF, ms = INF;
            #pragma unroll
            for (int j = 0; j < 4; ++j) {
                float dist = a2r + b2c[j] - 2.0f * acc[i][j][r];
                float dv   = (tar != tnc[j]) ? dist : 0.0f;          // d_valid (pre-inf)
                float dsv  = (dv > dapc[j] + TMARGIN) ? dv : 0.0f;   // d_semihard (pre-inf)
                mv = fminf(mv, (dv  == 0.0f) ? INF : dv);
                ms = fminf(ms, (dsv == 0.0f) ? INF : dsv);
            }
            #pragma unroll
            for (int off = 8; off; off >>= 1) {
                mv = fminf(mv, __shfl_xor(mv, off, 16));
                ms = fminf(ms, __shfl_xor(ms, off, 16));
            }
            if (l16 == 0) {
                atomicMin(&minValid[row], f2ord(mv));
                atomicMin(&minSemi[row],  f2ord(ms));
            }
        }
    }
}

__global__ __launch_bounds__(256) void tl_finalize(
    const float* __restrict__ dap, const unsigned* __restrict__ minSemi,
    const unsigned* __restrict__ minValid, float* __restrict__ out)
{
    __shared__ float red[256];
    float s = 0.0f;
    for (int i = threadIdx.x; i < NPTS; i += 256) {
        float ms = ord2f(minSemi[i]);
        float mv = ord2f(minValid[i]);
        float dan = isinf(ms) ? mv : ms;
        float l = dap[i] - dan + TMARGIN;   // relu(-inf) = 0 when no valid negative
        s += (l > 0.0f) ? l : 0.0f;
    }
    red[threadIdx.x] = s;
    __syncthreads();
    for (int stride = 128; stride; stride >>= 1) {
        if ((int)threadIdx.x < stride) red[threadIdx.x] += red[threadIdx.x + stride];
        __syncthreads();
    }
    if (threadIdx.x == 0) out[0] = red[0] / (float)NPTS;
}

extern "C" void kernel_launch(void* const* d_in, const int* in_sizes, int n_in,
                              void* d_out, int out_size, void* d_ws, size_t ws_size,
                              hipStream_t stream) {
    (void)in_sizes; (void)n_in; (void)out_size; (void)ws_size;
    const float* anchor   = (const float*)d_in[0];
    const float* positive = (const float*)d_in[1];
    const float* negative = (const float*)d_in[2];
    const int*   tA       = (const int*)d_in[3];
    const int*   tN       = (const int*)d_in[4];
    float* out = (float*)d_out;

    float*    a2       = (float*)d_ws;
    float*    b2       = a2 + NPTS;
    float*    dap      = b2 + NPTS;
    unsigned* minSemi  = (unsigned*)(dap + NPTS);
    unsigned* minValid = minSemi + NPTS;

    tl_prep<<<NPTS / 8, 256, 0, stream>>>(anchor, positive, negative,
                                          a2, b2, dap, minSemi, minValid);
    dim3 grid(NPTS / 128, NPTS / 256);
    tl_pairmin<<<grid, 256, LDS_BYTES, stream>>>(anchor, negative, a2, b2, dap,
                                                 tA, tN, minSemi, minValid);
    tl_finalize<<<1, 256, 0, stream>>>(dap, minSemi, minValid, out);
}